// CrfDecoderABC_46033459479310
// MI455X (gfx1250) — compile-verified
//
#include <hip/hip_runtime.h>

typedef float v2f __attribute__((ext_vector_type(2)));
typedef float v8f __attribute__((ext_vector_type(8)));

#define CRF_B 256
#define CRF_T 2048
#define CRF_N 64

// ---------------------------------------------------------------------------
// Kernel 1: gather path score per batch:
//   score[b] = start[tags[0]] + sum_t mask*em[b,t,tags[t]]
//            + sum_{t>=1} mask_t*Tr[tags[t-1],tags[t]] + end[tags[len-1]]
// ---------------------------------------------------------------------------
__global__ void crf_score_kernel(const float* __restrict__ em,
                                 const int* __restrict__ tags,
                                 const int* __restrict__ lengths,
                                 const float* __restrict__ trans,
                                 const float* __restrict__ start_t,
                                 const float* __restrict__ end_t,
                                 float* __restrict__ score_ws) {
  const int b   = blockIdx.x;
  const int tid = threadIdx.x;
  const int len = lengths[b];
  const int* tg = tags + b * CRF_T;
  const float* eb = em + (size_t)b * CRF_T * CRF_N;

  float acc = 0.f;
  for (int t = tid; t < CRF_T; t += 256) {
    if (t < len) {
      const int tt = tg[t];
      acc += eb[t * CRF_N + tt];
      if (t >= 1) acc += trans[tg[t - 1] * CRF_N + tt];
    }
  }
  __shared__ float red[256];
  red[tid] = acc;
  __syncthreads();
  for (int s = 128; s > 0; s >>= 1) {
    if (tid < s) red[tid] += red[tid + s];
    __syncthreads();
  }
  if (tid == 0) {
    score_ws[b] = red[0] + start_t[tg[0]] + end_t[tg[len - 1]];
  }
}

// ---------------------------------------------------------------------------
// Kernel 2: forward scan. 16 blocks x 128 threads (4 waves).
// Block handles 16 batch rows; wave w owns output columns [16w, 16w+16).
// Per step: rowmax -> P = exp(alpha - m) -> 16x WMMA_F32_16X16X4_F32
//           -> alpha = log(acc) + m + em (masked by t < length).
// ---------------------------------------------------------------------------
__launch_bounds__(128, 1)
__global__ void crf_forward_kernel(const float* __restrict__ em,
                                   const int* __restrict__ lengths,
                                   const float* __restrict__ trans,
                                   const float* __restrict__ start_t,
                                   const float* __restrict__ end_t,
                                   const float* __restrict__ score_ws,
                                   float* __restrict__ out) {
  __shared__ float expT[CRF_N][CRF_N];   // exp(transitions), 16 KB, constant
  __shared__ float alpha[16][CRF_N];     // running alpha, 4 KB
  __shared__ float P[16][CRF_N];         // exp(alpha - rowmax), 4 KB
  __shared__ float pmax[16][8];
  __shared__ float rmax[16];

  const int tid  = threadIdx.x;      // 0..127
  const int wave = tid >> 5;         // 0..3 : column tile
  const int lane = tid & 31;
  const int b0   = blockIdx.x * 16;  // base batch row

  // exp(transitions) into LDS.
  for (int i = tid; i < CRF_N * CRF_N; i += 128)
    expT[i >> 6][i & 63] = __expf(trans[i]);

  // alpha0 = start + emissions[:, 0]
  for (int i = tid; i < 16 * CRF_N; i += 128) {
    const int r = i >> 6, c = i & 63;
    alpha[r][c] = start_t[c] + em[(size_t)(b0 + r) * CRF_T * CRF_N + c];
  }
  __syncthreads();

  // --- Preload B fragments (constant all scan): 64x16 slab, 32 VGPRs/lane.
  // Assumed 32-bit B 4x16 layout: lane l -> N = l%16; VGPR v -> K = 2*(l/16)+v.
  const int bcol  = wave * 16 + (lane & 15);
  const int khalf = (lane >> 4) * 2;
  v2f bfrag[16];
#pragma unroll
  for (int kk = 0; kk < 16; ++kk) {
    bfrag[kk].x = expT[kk * 4 + khalf + 0][bcol];
    bfrag[kk].y = expT[kk * 4 + khalf + 1][bcol];
  }

  // Per-lane output mapping (C/D layout): M = v + 8*(lane/16), N = lane%16.
  const int arow  = lane & 15;
  const int ncol  = wave * 16 + (lane & 15);
  const int mbase = (lane >> 4) * 8;
  const float* emp[8];
  int mylen[8];
#pragma unroll
  for (int v = 0; v < 8; ++v) {
    const int gb = b0 + mbase + v;
    emp[v]   = em + (size_t)gb * CRF_T * CRF_N + ncol;
    mylen[v] = lengths[gb];
  }

  for (int t = 1; t < CRF_T; ++t) {
    // Row max, phase 1: each thread reduces 8 contiguous elements.
    {
      const int r = tid >> 3, seg = tid & 7;
      const float* ar = &alpha[r][seg * 8];
      float m = ar[0];
#pragma unroll
      for (int k = 1; k < 8; ++k) m = fmaxf(m, ar[k]);
      pmax[r][seg] = m;
    }
    __syncthreads();
    if (tid < 16) {
      float m = pmax[tid][0];
#pragma unroll
      for (int k = 1; k < 8; ++k) m = fmaxf(m, pmax[tid][k]);
      rmax[tid] = m;
    }
    __syncthreads();

    // P = exp(alpha - rowmax)
    for (int i = tid; i < 16 * CRF_N; i += 128) {
      const int r = i >> 6, c = i & 63;
      P[r][c] = __expf(alpha[r][c] - rmax[r]);
    }

    // Prefetch next step's emission rows (one lane per column tile half).
    if (((lane & 15) == 0) && (t + 1 < CRF_T)) {
#pragma unroll
      for (int v = 0; v < 8; ++v)
        __builtin_prefetch(emp[v] + (size_t)(t + 1) * CRF_N, 0, 1);
    }
    __syncthreads();

    // GEMM: acc(16x16) = P(16x64) @ expT(64 x this wave's 16 cols).
    // 4 independent accumulator chains to hide WMMA RAW latency.
    v8f acc0 = {}, acc1 = {}, acc2 = {}, acc3 = {};
#pragma unroll
    for (int kk = 0; kk < 16; kk += 4) {
      const v2f a0 = *(const v2f*)&P[arow][(kk + 0) * 4 + khalf];
      const v2f a1 = *(const v2f*)&P[arow][(kk + 1) * 4 + khalf];
      const v2f a2 = *(const v2f*)&P[arow][(kk + 2) * 4 + khalf];
      const v2f a3 = *(const v2f*)&P[arow][(kk + 3) * 4 + khalf];
      acc0 = __builtin_amdgcn_wmma_f32_16x16x4_f32(false, a0, false, bfrag[kk + 0],
                                                   (short)0, acc0, false, false);
      acc1 = __builtin_amdgcn_wmma_f32_16x16x4_f32(false, a1, false, bfrag[kk + 1],
                                                   (short)0, acc1, false, false);
      acc2 = __builtin_amdgcn_wmma_f32_16x16x4_f32(false, a2, false, bfrag[kk + 2],
                                                   (short)0, acc2, false, false);
      acc3 = __builtin_amdgcn_wmma_f32_16x16x4_f32(false, a3, false, bfrag[kk + 3],
                                                   (short)0, acc3, false, false);
    }
    const v8f acc = (acc0 + acc1) + (acc2 + acc3);

    // alpha_new = log(acc) + rowmax + em  (only where t < length)
#pragma unroll
    for (int v = 0; v < 8; ++v) {
      const int r = mbase + v;
      if (t < mylen[v]) {
        const float e = emp[v][(size_t)t * CRF_N];
        alpha[r][ncol] = __logf(acc[v]) + rmax[r] + e;
      }
    }
    __syncthreads();
  }

  // log_z = logsumexp_j(alpha + end);  out = score - log_z
  if (tid < 16) {
    const int r = tid, gb = b0 + r;
    float m = -1e30f;
    for (int j = 0; j < CRF_N; ++j) m = fmaxf(m, alpha[r][j] + end_t[j]);
    float s = 0.f;
    for (int j = 0; j < CRF_N; ++j) s += __expf(alpha[r][j] + end_t[j] - m);
    out[gb] = score_ws[gb] - (__logf(s) + m);
  }
}

// ---------------------------------------------------------------------------
extern "C" void kernel_launch(void* const* d_in, const int* in_sizes, int n_in,
                              void* d_out, int out_size, void* d_ws, size_t ws_size,
                              hipStream_t stream) {
  (void)in_sizes; (void)n_in; (void)out_size; (void)ws_size;
  const float* em      = (const float*)d_in[0];
  const int*   tags    = (const int*)d_in[1];    // jax int64 demotes to int32
  const int*   lengths = (const int*)d_in[2];
  const float* trans   = (const float*)d_in[3];
  const float* start_t = (const float*)d_in[4];
  const float* end_t   = (const float*)d_in[5];
  float* out      = (float*)d_out;
  float* score_ws = (float*)d_ws;   // 256 floats of scratch

  crf_score_kernel<<<CRF_B, 256, 0, stream>>>(em, tags, lengths, trans,
                                              start_t, end_t, score_ws);
  crf_forward_kernel<<<CRF_B / 16, 128, 0, stream>>>(em, lengths, trans,
                                                     start_t, end_t, score_ws, out);
}